// SubsetNeighborhoodSampler_32598801776713
// MI455X (gfx1250) — compile-verified
//
#include <hip/hip_runtime.h>
#include <hip/hip_bf16.h>
#include <stdint.h>
#include <stddef.h>

// Problem constants (match reference)
#define BN_TOTAL 16384      // B*N = 8*2048
#define M_DIM    4096
#define KSEL     16
#define NPTS     100
#define AVAL     5.0f
#define TPB      256        // 8 wave32 per block
#define NBLK     4096       // persistent blocks; 4 rows per block
#define NWAVES   (TPB/32)

typedef float v4f __attribute__((ext_vector_type(4)));

// Types matching the async-DMA builtin prototype: int __attribute__((vector_size(16)))
// with global (AS1) / LDS (AS3) address spaces.
typedef int v4i_vs __attribute__((vector_size(4 * sizeof(int))));
typedef __attribute__((address_space(1))) v4i_vs gbl_v4i;
typedef __attribute__((address_space(3))) v4i_vs lds_v4i;
typedef __attribute__((address_space(3))) void lds_void;

// ---------------- CDNA5 async global->LDS DMA (ASYNCcnt path) ----------------
__device__ __forceinline__ void async_copy_b128(const float* g, float* l) {
#if __has_builtin(__builtin_amdgcn_global_load_async_to_lds_b128)
  __builtin_amdgcn_global_load_async_to_lds_b128((gbl_v4i*)g, (lds_v4i*)l, 0, 0);
#else
  unsigned laddr = (unsigned)(unsigned long long)(lds_void*)l;
  asm volatile("global_load_async_to_lds_b128 %0, %1, off"
               :: "v"(laddr), "v"((unsigned long long)g)
               : "memory");
#endif
}

#if __has_builtin(__builtin_amdgcn_s_wait_asynccnt)
#define WAIT_ASYNC(n) __builtin_amdgcn_s_wait_asynccnt(n)
#else
#define WAIT_ASYNC(n) asm volatile("s_wait_asynccnt %0" :: "n"(n) : "memory")
#endif

// ---------------- helpers ----------------
__device__ __forceinline__ unsigned float_key(float f) {
  // order-preserving float -> u32 (works for negatives / -inf)
  unsigned u = __float_as_uint(f);
  return (u & 0x80000000u) ? ~u : (u | 0x80000000u);
}

__device__ __forceinline__ float log1mexp(float x) {
  // log(1 - exp(-x)) for x > 0, stable on both ends (matches reference)
  x = fmaxf(x, 1e-30f);
  return (x > 0.693f) ? log1pf(-expf(-x)) : logf(-expm1f(-x));
}

// butterfly all-reduce max over one wave32; every lane gets the result
__device__ __forceinline__ unsigned long long wave_allmax_u64(unsigned long long key) {
  #pragma unroll
  for (int m = 16; m > 0; m >>= 1) {
    unsigned long long o = __shfl_xor(key, m, 32);
    if (o > key) key = o;
  }
  return key;
}

__device__ __forceinline__ float block_max_f32(float v, float* red, int tid) {
  #pragma unroll
  for (int off = 16; off > 0; off >>= 1) v = fmaxf(v, __shfl_down(v, (unsigned)off, 32));
  if ((tid & 31) == 0) red[tid >> 5] = v;
  __syncthreads();
  if (tid == 0) {
    float b = red[0];
    #pragma unroll
    for (int w = 1; w < NWAVES; ++w) b = fmaxf(b, red[w]);
    red[NWAVES] = b;
  }
  __syncthreads();
  return red[NWAVES];
}

__device__ __forceinline__ float block_sum_f32(float v, float* red, int tid) {
  #pragma unroll
  for (int off = 16; off > 0; off >>= 1) v += __shfl_down(v, (unsigned)off, 32);
  if ((tid & 31) == 0) red[tid >> 5] = v;
  __syncthreads();
  if (tid == 0) {
    float b = 0.0f;
    #pragma unroll
    for (int w = 0; w < NWAVES; ++w) b += red[w];
    red[NWAVES] = b;
  }
  __syncthreads();
  return red[NWAVES];
}

// stage one row (scores + gumbel, 32 KB) into LDS via async DMA: 8 ops/wave
__device__ __forceinline__ void stage_row(const float* gs, const float* gg,
                                          float* ls, float* lg, int tid) {
  #pragma unroll
  for (int j = 0; j < 4; ++j) {
    int v = j * TPB + tid;
    async_copy_b128(gs + v * 4, ls + v * 4);
    async_copy_b128(gg + v * 4, lg + v * 4);
  }
}

// ---------------- kernel ----------------
__global__ void __launch_bounds__(TPB)
subset_sampler_kernel(const float* __restrict__ scores,
                      const float* __restrict__ gumbel,
                      const float* __restrict__ taup,
                      float* __restrict__ out) {
  __shared__ __align__(16) float bufS[2][M_DIM];
  __shared__ __align__(16) float bufG[2][M_DIM];
  __shared__ unsigned long long waveCand[NWAVES * KSEL];   // 8 waves x 16 candidates
  __shared__ unsigned long long selKey[KSEL];              // final top-16 keys
  __shared__ float redF[NWAVES + 1];
  __shared__ float selPhi[KSEL];

  const int tid  = threadIdx.x;
  const int lane = tid & 31;
  const int wave = tid >> 5;
  const float tau = taup[0];
  float* ll_out = out + (size_t)BN_TOTAL * M_DIM;

  int row = blockIdx.x;
  int stg = 0;
  if (row < BN_TOTAL)
    stage_row(scores + (size_t)row * M_DIM, gumbel + (size_t)row * M_DIM,
              bufS[0], bufG[0], tid);

  for (; row < BN_TOTAL; row += gridDim.x) {
    const int nxt = row + (int)gridDim.x;
    const bool hasn = nxt < BN_TOTAL;
    if (hasn)
      stage_row(scores + (size_t)nxt * M_DIM, gumbel + (size_t)nxt * M_DIM,
                bufS[stg ^ 1], bufG[stg ^ 1], tid);
    if (hasn) { WAIT_ASYNC(8); } else { WAIT_ASYNC(0); }   // current row's 8 DMA ops done
    __syncthreads();                                        // ... for every wave

    // ---- LDS -> registers: phi = scores/tau, sample = phi + gumbel ----
    const v4f* bs = (const v4f*)bufS[stg];
    const v4f* bg = (const v4f*)bufG[stg];
    float phi_r[16], samp_r[16];
    #pragma unroll
    for (int j = 0; j < 4; ++j) {
      v4f sv = bs[j * TPB + tid];
      v4f gv = bg[j * TPB + tid];
      #pragma unroll
      for (int c = 0; c < 4; ++c) {
        float ph = sv[c] / tau;
        phi_r[4 * j + c]  = ph;
        samp_r[4 * j + c] = ph + gv[c];
      }
    }

    // ---- Phase 1: per-wave top-16 of its 512 elements (shuffle-only) ----
    #pragma unroll 1
    for (int it = 0; it < KSEL; ++it) {
      float best = -__builtin_inff(); int be = 0;
      #pragma unroll
      for (int e = 0; e < 16; ++e)
        if (samp_r[e] > best) { best = samp_r[e]; be = e; }
      // global column: col = 1024*j + 4*tid + c
      int col = ((be >> 2) << 10) + (tid << 2) + (be & 3);
      unsigned long long key =
          ((unsigned long long)float_key(best) << 32) | (unsigned)col;
      unsigned long long win = wave_allmax_u64(key);
      int wcol = (int)(unsigned)(win & 0xffffffffull);
      if (((wcol >> 2) & (TPB - 1)) == tid) {          // owner lane masks it out
        int e = ((wcol >> 10) << 2) | (wcol & 3);
        samp_r[e] = -__builtin_inff();
      }
      if (lane == 0) waveCand[wave * KSEL + it] = win;
    }
    __syncthreads();

    // ---- Phase 2: wave 0 merges 128 candidates -> global top-16 (shuffle-only) ----
    if (wave == 0) {
      unsigned long long c0 = waveCand[lane];
      unsigned long long c1 = waveCand[lane + 32];
      unsigned long long c2 = waveCand[lane + 64];
      unsigned long long c3 = waveCand[lane + 96];
      #pragma unroll 1
      for (int it = 0; it < KSEL; ++it) {
        unsigned long long b01 = c0 > c1 ? c0 : c1;
        unsigned long long b23 = c2 > c3 ? c2 : c3;
        unsigned long long best = b01 > b23 ? b01 : b23;
        unsigned long long win = wave_allmax_u64(best);
        // keys are unique (column in low bits) -> exactly one lane masks
        if (c0 == win) c0 = 0ull;
        if (c1 == win) c1 = 0ull;
        if (c2 == win) c2 = 0ull;
        if (c3 == win) c3 = 0ull;
        if (lane == 0) selKey[it] = win;
      }
    }
    __syncthreads();

    // ---- Phase 3: selection mask + selected phi from winning columns ----
    unsigned selmask = 0u;
    #pragma unroll
    for (int it = 0; it < KSEL; ++it) {
      int col = (int)(unsigned)(selKey[it] & 0xffffffffull);
      if (((col >> 2) & (TPB - 1)) == tid) {
        int e = ((col >> 10) << 2) | (col & 3);
        selmask |= 1u << e;
        selPhi[it] = phi_r[e];
      }
    }
    __syncthreads();   // selPhi visible to all

    // ---- khot row: write-once 268MB stream -> non-temporal b128 stores ----
    {
      v4f* orow = (v4f*)(out + (size_t)row * M_DIM);
      #pragma unroll
      for (int j = 0; j < 4; ++j) {
        v4f o;
        #pragma unroll
        for (int c = 0; c < 4; ++c)
          o[c] = ((selmask >> (4 * j + c)) & 1u) ? 1.0f : 0.0f;
        __builtin_nontemporal_store(o, orow + j * TPB + tid);
      }
    }

    // ---- log-softmax normalizer over all M=4096 ----
    float lmax = -__builtin_inff();
    #pragma unroll
    for (int e = 0; e < 16; ++e) lmax = fmaxf(lmax, phi_r[e]);
    lmax = block_max_f32(lmax, redF, tid);
    float sume = 0.0f;
    #pragma unroll
    for (int e = 0; e < 16; ++e) sume += expf(phi_r[e] - lmax);
    sume = block_sum_f32(sume, redF, tid);
    const float logZ = lmax + logf(sume);

    // ---- selected probabilities (broadcast via LDS) ----
    float p[KSEL]; float sump = 0.0f;
    #pragma unroll
    for (int k = 0; k < KSEL; ++k) {
      p[k] = expf(selPhi[k] - logZ);
      sump += p[k];
    }
    const float w_c = fmaxf(1.0f - sump, 1e-12f);

    // ---- 100-point trapezoid integration, one point per thread ----
    float li = -__builtin_inff();
    if (tid < NPTS) {
      const float dstep = 2.0f * AVAL / (float)(NPTS - 1);
      float sv = -AVAL + dstep * (float)tid;
      float t  = expf(sv);
      float lw = logf(dstep) + ((tid == 0 || tid == NPTS - 1) ? logf(0.5f) : 0.0f);
      float acc = 0.0f;
      #pragma unroll
      for (int k = 0; k < KSEL; ++k) acc += log1mexp(p[k] * t);
      li = logf(w_c) - w_c * t + sv + acc + lw;
    }
    float m  = block_max_f32(li, redF, tid);
    float se = block_sum_f32((tid < NPTS) ? expf(li - m) : 0.0f, redF, tid);
    if (tid == 0) ll_out[row] = m + logf(se);

    __syncthreads();   // all reads of bufS/bufG[stg], selKey, selPhi done before reuse
    stg ^= 1;
  }
}

// ---------------- launcher ----------------
extern "C" void kernel_launch(void* const* d_in, const int* in_sizes, int n_in,
                              void* d_out, int out_size, void* d_ws, size_t ws_size,
                              hipStream_t stream) {
  (void)in_sizes; (void)n_in; (void)out_size; (void)d_ws; (void)ws_size;
  const float* scores = (const float*)d_in[0];
  const float* gumbel = (const float*)d_in[1];
  const float* tau    = (const float*)d_in[2];
  float* out = (float*)d_out;
  subset_sampler_kernel<<<dim3(NBLK), dim3(TPB), 0, stream>>>(scores, gumbel, tau, out);
}